// DBNDownBeatTrackingProcessor_31739808317569
// MI455X (gfx1250) — compile-verified
//
#include <hip/hip_runtime.h>
#include <cstdint>
#include <cstddef>

// DBN downbeat Viterbi, MI455X (gfx1250).
// Single persistent workgroup, LDS-resident state, TDM-staged constants,
// compressed backpointers (240 uint16 per frame) in d_ws (needs T*240*2 =
// 2.88 MB; sits in L2).

#define NTHREADS 1024
#define NTEMPI   60
#define NBEATS   4
#define NPAIRS   (NBEATS * NTEMPI)            /* 240 */
#define TRANS_N  (NBEATS * NTEMPI * NTEMPI)   /* 14400 */
#define MAXCH    20
#define NEGF     (-1e30f)

typedef __attribute__((ext_vector_type(4))) unsigned int u32x4;
typedef __attribute__((ext_vector_type(8))) unsigned int u32x8;

__device__ __forceinline__ unsigned lds_addr_of(const void* p) {
  // generic -> LDS(3) addrspace cast gives the raw LDS byte offset
  return (unsigned)(uintptr_t)(__attribute__((address_space(3))) const char*)p;
}

// TDM: DMA `ndw` dwords from global to LDS. 1-row 2D tile, data_size=4B.
// D# layout per CDNA5 ISA ch.8 (group0 128b, group1 256b); VADDR2/3 NULL
// (short form) since tensor is <=2D.
__device__ __forceinline__ void tdm_load_dwords(const void* lds_dst,
                                                const void* gsrc,
                                                unsigned ndw) {
  unsigned long long ga = (unsigned long long)(uintptr_t)gsrc;
  u32x4 g0;
  g0[0] = 1u;                                                   // count=1
  g0[1] = lds_addr_of(lds_dst);                                 // lds_addr
  g0[2] = (unsigned)(ga & 0xFFFFFFFFull);                       // gaddr[31:0]
  g0[3] = (unsigned)((ga >> 32) & 0x01FFFFFFull) | (2u << 30);  // gaddr[56:32]|type=2
  u32x8 g1;
  g1[0] = (2u << 16);                // wg_mask=0, data_size=2 (4 bytes)
  g1[1] = (ndw & 0xFFFFu) << 16;     // tensor_dim0[15:0]  (bits 63:48)
  g1[2] = (ndw >> 16) | (1u << 16);  // tensor_dim0[31:16], tensor_dim1=1
  g1[3] = (ndw & 0xFFFFu) << 16;     // tensor_dim1 hi=0, tile_dim0=ndw
  g1[4] = 1u;                        // tile_dim1=1, tile_dim2=0
  g1[5] = ndw;                       // tensor_dim0_stride[31:0]
  g1[6] = 0u;                        // stride hi, dim1_stride lo
  g1[7] = 0u;
  asm volatile("tensor_load_to_lds %0, %1" :: "s"(g0), "s"(g1) : "memory");
}

extern "C" __global__ void __launch_bounds__(NTHREADS)
dbn_viterbi_kernel(const float* __restrict__ acts,        // (T,2)
                   const float* __restrict__ log_trans_g, // (4,60,60)
                   const int*   __restrict__ prev_last_g, // (4,60)
                   const int*   __restrict__ first_states_g, // (4,60)
                   const int*   __restrict__ pointer_g,   // (S,)
                   float*       __restrict__ out,         // T path + 1 logp
                   unsigned short* __restrict__ bp,       // (T,240) in d_ws
                   int S, int T) {
  extern __shared__ char smem[];
  const int tid = threadIdx.x;

  // ---- LDS layout (must match host shared-size formula) ----
  char* b8 = smem;
  float* v     = (float*)b8;  b8 += 4 * (size_t)S;        // Viterbi values
  float* trans = (float*)b8;  b8 += 4 * (size_t)TRANS_N;  // log transitions
  float* actsl = (float*)b8;  b8 += 4 * (size_t)(2 * T);  // all activations
  int*   prevl = (int*)b8;    b8 += 4 * NPAIRS;
  float* bestl = (float*)b8;  b8 += 4 * NPAIRS;
  float* redv  = (float*)b8;  b8 += 4 * NTHREADS;
  int*   redi  = (int*)b8;    b8 += 4 * NTHREADS;
  float* dens3 = (float*)b8;  b8 += 16;
  unsigned char* ptrb = (unsigned char*)b8; b8 += (size_t)S;  // obs index 0/1/2
  unsigned char* fidx = (unsigned char*)b8;                   // 255 or pair id

  // ---- init: TDM-stage big constant blocks, build tables in parallel ----
  if (tid == 0) {
    tdm_load_dwords(trans, log_trans_g, (unsigned)TRANS_N);
    tdm_load_dwords(actsl, acts, (unsigned)(2 * T));
  }
  __builtin_amdgcn_s_wait_tensorcnt(0);   // wave0 retires its DMAs; others no-op

  const float v0 = -logf((float)S);
  for (int s = tid; s < S; s += NTHREADS) {
    ptrb[s] = (unsigned char)pointer_g[s];
    fidx[s] = 255u;
    v[s]    = v0;
  }
  if (tid < NPAIRS) prevl[tid] = prev_last_g[tid];
  __syncthreads();
  if (tid < NPAIRS) fidx[first_states_g[tid]] = (unsigned char)tid;

  // chunk assignment (contiguous per thread -> exact numpy argmax tie-break)
  int CH = (S + NTHREADS - 1) / NTHREADS;
  if (CH > MAXCH) CH = MAXCH;             // S is ~14.5K; never triggers
  const int c0 = tid * CH;

  const int p   = tid >> 2;               // pair id (b*60+j), 4 lanes/pair
  const int sub = tid & 3;

  // ---- forward pass: 2 barriers / frame ----
  for (int t = 0; t < T; ++t) {
    __syncthreads();                      // v from t-1 fully written

    // Stage A: best[b,j] = max_i v[prev_last[b,i]] + log_trans[b,i,j]
    if (p < NPAIRS) {
      const int b = p / NTEMPI;
      const int j = p - b * NTEMPI;
      float bv = -3.4e38f; int ba = 0;
      const int i0 = sub * (NTEMPI / 4);
#pragma unroll
      for (int k = 0; k < NTEMPI / 4; ++k) {
        const int i = i0 + k;
        const float c = v[prevl[b * NTEMPI + i]] +
                        trans[(b * NTEMPI + i) * NTEMPI + j];
        if (c > bv) { bv = c; ba = i; }   // keeps first max within chunk
      }
#pragma unroll
      for (int m = 1; m <= 2; m <<= 1) {  // combine 4 lanes (wave32 shuffles)
        const float ov = __shfl_xor(bv, m);
        const int   oa = __shfl_xor(ba, m);
        if (ov > bv || (ov == bv && oa < ba)) { bv = ov; ba = oa; }
      }
      if (sub == 0) {
        bestl[p] = bv;
        bp[(size_t)t * NPAIRS + p] =
            (unsigned short)prevl[(p / NTEMPI) * NTEMPI + ba];
      }
    } else if (tid == NPAIRS * 4) {       // thread 960: observation densities
      const float ab = actsl[2 * t], ad = actsl[2 * t + 1];
      dens3[0] = logf((1.0f - ab - ad) * (1.0f / 15.0f)); // OBS_LAMBDA-1 = 15
      dens3[1] = logf(ab);
      dens3[2] = logf(ad);
    }

    // Stage B read: in-place shift -> stage v[s-1] in registers
    float r[MAXCH];
#pragma unroll
    for (int k = 0; k < MAXCH; ++k) {
      const int s = c0 + k;
      if (k < CH && s < S) r[k] = (s > 0) ? v[s - 1] : NEGF;
    }
    __syncthreads();                      // A results ready, reads done

    // Stage B write: v[s] = (first? best : v[s-1]) + dens[ptr[s]]
#pragma unroll
    for (int k = 0; k < MAXCH; ++k) {
      const int s = c0 + k;
      if (k < CH && s < S) {
        const unsigned f = fidx[s];
        const float val = (f != 255u) ? bestl[f] : r[k];
        v[s] = val + dens3[ptrb[s]];
      }
    }
  }

  __threadfence();                        // flush bp stores toward L2
  __syncthreads();

  // ---- argmax over final v (first-max semantics) ----
  float mv = -3.4e38f; int mi = S;
  for (int k = 0; k < CH; ++k) {
    const int s = c0 + k;
    if (s < S && v[s] > mv) { mv = v[s]; mi = s; }
  }
  redv[tid] = mv; redi[tid] = mi;
  __syncthreads();
  for (int off = NTHREADS / 2; off > 0; off >>= 1) {
    if (tid < off) {
      const float ov = redv[tid + off]; const int oi = redi[tid + off];
      if (ov > redv[tid] || (ov == redv[tid] && oi < redi[tid])) {
        redv[tid] = ov; redi[tid] = oi;
      }
    }
    __syncthreads();
  }

  // ---- backtrack (inherently serial; LDS byte lookup per step, global
  //      load only at beat boundaries ~T/55 times) ----
  if (tid == 0) {
    int s = redi[0];
    const float logp = redv[0];
    out[T - 1] = (float)s;
    for (int t = T - 1; t >= 1; --t) {
      const unsigned f = fidx[s];
      const int prv = (f != 255u) ? (int)bp[(size_t)t * NPAIRS + f] : (s - 1);
      out[t - 1] = (float)prv;
      s = prv;
    }
    out[T] = logp;
  }
}

extern "C" void kernel_launch(void* const* d_in, const int* in_sizes, int n_in,
                              void* d_out, int out_size, void* d_ws,
                              size_t ws_size, hipStream_t stream) {
  const float* acts         = (const float*)d_in[0];
  const float* log_trans    = (const float*)d_in[1];
  const int*   prev_last    = (const int*)d_in[2];
  const int*   first_states = (const int*)d_in[3];
  const int*   pointer      = (const int*)d_in[4];
  // d_in[5] = num_states scalar; sizes give it directly:
  const int S = in_sizes[4];       // pointer array length == num_states
  const int T = in_sizes[0] / 2;   // activations are (T,2)

  // must match device LDS layout: 4S + 57600 + 8T + 960+960+4096+4096+16 + 2S
  const size_t shbytes = (size_t)6 * S + (size_t)8 * T + 67728;
  (void)hipFuncSetAttribute((const void*)dbn_viterbi_kernel,
                            hipFuncAttributeMaxDynamicSharedMemorySize,
                            (int)shbytes);

  dbn_viterbi_kernel<<<dim3(1), dim3(NTHREADS), shbytes, stream>>>(
      acts, log_trans, prev_last, first_states, pointer,
      (float*)d_out, (unsigned short*)d_ws, S, T);
}